// TripletLoss_2078764171739
// MI455X (gfx1250) — compile-verified
//
#include <hip/hip_runtime.h>
#include <math.h>

typedef float v2f __attribute__((ext_vector_type(2)));
typedef float v8f __attribute__((ext_vector_type(8)));

#define N_ROWS 8192
#define DIMK   128
#define TMARGIN 0.3f

// ---------------------------------------------------------------------------
// Kernel 1: per-row squared norms. One wave (32 lanes) per row; each lane
// loads one float4 (128 floats / row) and we shfl-reduce within the wave.
// ---------------------------------------------------------------------------
__global__ __launch_bounds__(256) void sqnorm_kernel(const float* __restrict__ x,
                                                     float* __restrict__ sq) {
    int wave = (blockIdx.x * blockDim.x + threadIdx.x) >> 5;
    int lane = threadIdx.x & 31;
    const float4* row = (const float4*)(x + (size_t)wave * DIMK);
    float4 v = row[lane];
    float s = v.x * v.x + v.y * v.y + v.z * v.z + v.w * v.w;
    #pragma unroll
    for (int m = 16; m >= 1; m >>= 1) s += __shfl_xor(s, m, 32);
    if (lane == 0) sq[wave] = s;
}

// ---------------------------------------------------------------------------
// Kernel 2: tiled Gram matrix via V_WMMA_F32_16X16X4_F32 + fused batch-hard
// mining. 64 workgroups x 256 threads (8 waves). Wave w owns rows
// [blk*128 + 16w, +16): its A panel (16x128 f32) lives in 32 v2f registers
// per lane for the entire kernel. Column blocks of 64 rows are staged in LDS.
// ---------------------------------------------------------------------------
__global__ __launch_bounds__(256) void triplet_tile_kernel(
    const float* __restrict__ x, const int* __restrict__ t,
    const float* __restrict__ sq,
    float* __restrict__ apOut, float* __restrict__ anOut)
{
    __shared__ float Bs[64 * DIMK];                 // 32 KB column-block stage

    const int tid      = threadIdx.x;
    const int lane     = tid & 31;
    const int wave     = tid >> 5;
    const int rBase    = blockIdx.x * 128 + wave * 16;
    const int laneHalf = lane >> 4;                 // 0 or 1
    const int lmod     = lane & 15;
    const int kOff     = laneHalf * 2;              // K sub-offset per ISA layout

    // --- A panel into registers (32-bit A 16x4 layout, 32 K-chunks of 4) ---
    v2f a[32];
    const float* arow = x + (size_t)(rBase + lmod) * DIMK + kOff;
    #pragma unroll
    for (int kk = 0; kk < 32; ++kk)
        a[kk] = *(const v2f*)(arow + kk * 4);

    // --- per-lane row metadata following C/D layout: VGPR m <-> row m + 8*half
    float rowSq[8]; int rowT[8];
    #pragma unroll
    for (int m = 0; m < 8; ++m) {
        int r = rBase + m + laneHalf * 8;
        rowSq[m] = sq[r];
        rowT[m]  = t[r];
    }

    float ap[8], an[8];
    #pragma unroll
    for (int m = 0; m < 8; ++m) { ap[m] = -INFINITY; an[m] = INFINITY; }

    for (int cb = 0; cb < N_ROWS / 64; ++cb) {
        const int colBlock = cb * 64;
        __syncthreads();
        // cooperative load: 64 rows x 128 f32 = 8192 floats = 2048 float4
        const float4* src = (const float4*)(x + (size_t)colBlock * DIMK);
        float4*       dst = (float4*)Bs;
        #pragma unroll
        for (int i = 0; i < 8; ++i)
            dst[i * 256 + tid] = src[i * 256 + tid];
        __syncthreads();

        #pragma unroll 1
        for (int n4 = 0; n4 < 4; ++n4) {
            const int   colBase = colBlock + n4 * 16;
            const int   col     = colBase + lmod;     // this lane's N column
            const float colSq   = sq[col];
            const int   colT    = t[col];

            // B 4x16 layout: lane = N, halves hold K={0,1}/{2,3} of each chunk
            const float* brow = Bs + (n4 * 16 + lmod) * DIMK + kOff;

            v8f acc = {};
            #pragma unroll
            for (int kk = 0; kk < 32; ++kk) {
                v2f b = *(const v2f*)(brow + kk * 4);
                acc = __builtin_amdgcn_wmma_f32_16x16x4_f32(
                    /*neg_a=*/false, a[kk], /*neg_b=*/false, b,
                    /*c_mod=*/(short)0, acc, /*reuse_a=*/false, /*reuse_b=*/false);
            }

            #pragma unroll
            for (int m = 0; m < 8; ++m) {
                float d2   = rowSq[m] + colSq - 2.0f * acc[m];
                float d    = sqrtf(fmaxf(d2, 1e-12f));
                bool  same = (rowT[m] == colT);
                ap[m] = fmaxf(ap[m], same ? d : -INFINITY);
                an[m] = fminf(an[m], same ? INFINITY : d);
            }
        }
    }

    // reduce across the 16 lanes sharing each row (xor masks <16 stay in-half)
    #pragma unroll
    for (int m = 0; m < 8; ++m) {
        #pragma unroll
        for (int s = 8; s >= 1; s >>= 1) {
            ap[m] = fmaxf(ap[m], __shfl_xor(ap[m], s, 32));
            an[m] = fminf(an[m], __shfl_xor(an[m], s, 32));
        }
    }
    if (lmod == 0) {
        #pragma unroll
        for (int m = 0; m < 8; ++m) {
            int r = rBase + m + laneHalf * 8;
            apOut[r] = ap[m];
            anOut[r] = an[m];
        }
    }
}

// ---------------------------------------------------------------------------
// Kernel 3: final scalar reduction -> loss, precision
// ---------------------------------------------------------------------------
__global__ __launch_bounds__(256) void finalize_kernel(const float* __restrict__ ap,
                                                       const float* __restrict__ an,
                                                       float* __restrict__ out) {
    __shared__ float lossBuf[256];
    __shared__ float precBuf[256];
    int tid = threadIdx.x;
    float lossSum = 0.f, precSum = 0.f;
    for (int i = tid; i < N_ROWS; i += 256) {
        float a = ap[i], b = an[i];
        lossSum += fmaxf(0.f, TMARGIN - (b - a));
        precSum += (b > a) ? 1.f : 0.f;
    }
    lossBuf[tid] = lossSum;
    precBuf[tid] = precSum;
    __syncthreads();
    #pragma unroll
    for (int s = 128; s >= 1; s >>= 1) {
        if (tid < s) {
            lossBuf[tid] += lossBuf[tid + s];
            precBuf[tid] += precBuf[tid + s];
        }
        __syncthreads();
    }
    if (tid == 0) {
        out[0] = lossBuf[0] * (1.0f / N_ROWS);
        out[1] = precBuf[0] * (1.0f / N_ROWS);
    }
}

// ---------------------------------------------------------------------------
extern "C" void kernel_launch(void* const* d_in, const int* in_sizes, int n_in,
                              void* d_out, int out_size, void* d_ws, size_t ws_size,
                              hipStream_t stream) {
    const float* x = (const float*)d_in[0];
    const int*   t = (const int*)d_in[1];

    float* sq = (float*)d_ws;            // 8192 f32
    float* ap = sq + N_ROWS;             // 8192 f32
    float* an = ap + N_ROWS;             // 8192 f32
    float* out = (float*)d_out;          // [loss, prec]

    sqnorm_kernel<<<N_ROWS / 8, 256, 0, stream>>>(x, sq);
    triplet_tile_kernel<<<N_ROWS / 128, 256, 0, stream>>>(x, t, sq, ap, an);
    finalize_kernel<<<1, 256, 0, stream>>>(ap, an, out);
}